// TtMambaSSM_77661598646505
// MI455X (gfx1250) — compile-verified
//
#include <hip/hip_runtime.h>

// out = abar * hidden  (elementwise, fp32) — HBM-streaming kernel.
// 12 B/element, zero reuse -> non-temporal 128-bit loads/stores.
// Wave-uniform bounds branch: full blocks run a branchless fast path
// (no exec masking, no reg zero-init); only the last block guards.

typedef float v4f __attribute__((ext_vector_type(4)));

#define TPB 256        // 8 wave32 per block
#define VPT 4          // float4s per thread
#define PER_BLOCK (TPB * VPT)

__global__ __launch_bounds__(TPB) void TtMambaSSM_decay_mul_kernel(
    const float* __restrict__ abar,
    const float* __restrict__ hidden,
    float* __restrict__ out,
    int n4,   // number of float4 elements
    int n)    // total scalar elements
{
    const v4f* __restrict__ a4 = reinterpret_cast<const v4f*>(abar);
    const v4f* __restrict__ h4 = reinterpret_cast<const v4f*>(hidden);
    v4f* __restrict__ o4 = reinterpret_cast<v4f*>(out);

    const int base = blockIdx.x * PER_BLOCK + threadIdx.x;
    const bool last_block = (blockIdx.x == gridDim.x - 1);  // scalar condition

    if (!last_block) {
        // Fast path: whole block's span is in range. Unconditional loads ->
        // clause of 8 b128 NT loads per lane, then 16 muls, 4 NT stores.
        v4f va[VPT], vh[VPT];
#pragma unroll
        for (int j = 0; j < VPT; ++j) {
            va[j] = __builtin_nontemporal_load(&a4[base + j * TPB]);
            vh[j] = __builtin_nontemporal_load(&h4[base + j * TPB]);
        }
#pragma unroll
        for (int j = 0; j < VPT; ++j) {
            __builtin_nontemporal_store(va[j] * vh[j], &o4[base + j * TPB]);
        }
    } else {
        // Slow path: only the final block. Per-lane guards + scalar tail.
#pragma unroll
        for (int j = 0; j < VPT; ++j) {
            int i = base + j * TPB;
            if (i < n4) {
                v4f va = __builtin_nontemporal_load(&a4[i]);
                v4f vh = __builtin_nontemporal_load(&h4[i]);
                __builtin_nontemporal_store(va * vh, &o4[i]);
            }
        }
        // Scalar remainder (n % 4 != 0). Never taken for the ref shape.
        int t = n4 * 4 + threadIdx.x;
        if (t < n) {
            out[t] = abar[t] * hidden[t];
        }
    }
}

extern "C" void kernel_launch(void* const* d_in, const int* in_sizes, int n_in,
                              void* d_out, int out_size, void* d_ws, size_t ws_size,
                              hipStream_t stream) {
    (void)n_in; (void)out_size; (void)d_ws; (void)ws_size;

    const float* abar   = (const float*)d_in[0];
    const float* hidden = (const float*)d_in[1];
    float* out = (float*)d_out;

    int n  = in_sizes[0];      // 1*1*1024*32768 = 33,554,432 fp32 elements
    int n4 = n >> 2;           // float4 count

    int blocks = (n4 + PER_BLOCK - 1) / PER_BLOCK;  // 8192 for the ref shape
    if (blocks < 1) blocks = 1;

    TtMambaSSM_decay_mul_kernel<<<blocks, TPB, 0, stream>>>(abar, hidden, out, n4, n);
}